// Retinanet_37538014167635
// MI455X (gfx1250) — compile-verified
//
#include <hip/hip_runtime.h>
#include <math.h>
#include <stdint.h>

// ---------------- problem constants (match reference) ----------------
#define Bn 2
#define Nn 120000
#define Cn 80
#define Kk 300
#define MAXDET 100
#define CAPC 512                 // per-(b,c) candidate capacity (avg ~150 pass)
#define SCORE_THRES 0.99875f
#define NMS_THRES 0.5f
#define MIN_SIZE 0.001f
#define BBOX_CLIP 4.135166556742356f   // log(1000/16)
#define KPAD 304                 // 300 rounded up to 16
#define NWORDS 10                // ceil(304/32)
#define VWORDS 3750              // ceil(120000/32)

typedef __attribute__((ext_vector_type(2))) float v2f;
typedef __attribute__((ext_vector_type(8))) float v8f;

// ---- async global->LDS helpers (CDNA5 ASYNCcnt path, ISA 08 §4) ----
__device__ __forceinline__ void async_g2l_b32(void* lds, const void* gaddr) {
    unsigned l = (unsigned)(uintptr_t)lds;               // LDS byte offset
    unsigned long long g = (unsigned long long)(uintptr_t)gaddr;
    asm volatile("global_load_async_to_lds_b32 %0, %1, off"
                 :: "v"(l), "v"(g) : "memory");
}
__device__ __forceinline__ void async_g2l_b128(void* lds, const void* gaddr) {
    unsigned l = (unsigned)(uintptr_t)lds;
    unsigned long long g = (unsigned long long)(uintptr_t)gaddr;
    asm volatile("global_load_async_to_lds_b128 %0, %1, off"
                 :: "v"(l), "v"(g) : "memory");
}
__device__ __forceinline__ void wait_async0() {
    asm volatile("s_wait_asynccnt 0x0" ::: "memory");
}

// ---------------- kernel 0: zero per-(b,c) candidate counters ----------------
__global__ void k_zero_counts(int* counts) {
    int i = blockIdx.x * blockDim.x + threadIdx.x;
    if (i < Bn * Cn) counts[i] = 0;
}

// ---------------- kernel 1: streaming threshold scan + decode + compact ------
// One thread per (b, n, 4-class chunk): pure coalesced float4 stream of
// cls_preds (76.8 MB -> ~3.5us at 23.3 TB/s); survivors (~0.125%) decode the
// box once and push (score, anchor idx, decoded box) into per-(b,c) lists.
__global__ void k_gather(const float* __restrict__ boxp,
                         const float* __restrict__ clsp,
                         const float* __restrict__ anch,
                         const int* __restrict__ Hs, const int* __restrict__ Ws,
                         int* __restrict__ counts,
                         float* __restrict__ cscore,
                         int* __restrict__ cidx,
                         float* __restrict__ cbox) {
    const long total = (long)Bn * Nn * (Cn / 4);
    long tid = (long)blockIdx.x * blockDim.x + threadIdx.x;
    if (tid >= total) return;

    int  q  = (int)(tid % (Cn / 4));
    long bn = tid / (Cn / 4);
    int  n  = (int)(bn % Nn);
    int  b  = (int)(bn / Nn);

    const float4* cls4 = (const float4*)clsp;
    // keep the HBM stream ahead of us (global_prefetch_b8)
    __builtin_prefetch(cls4 + (bn * (Cn / 4) + q) + 4096, 0, 0);

    float4 s4 = cls4[bn * (Cn / 4) + q];
    bool p0 = s4.x > SCORE_THRES, p1 = s4.y > SCORE_THRES;
    bool p2 = s4.z > SCORE_THRES, p3 = s4.w > SCORE_THRES;
    if (!(p0 | p1 | p2 | p3)) return;   // common path: 16B read, exit

    float W = (float)Ws[0], H = (float)Hs[0];
    float4 d = ((const float4*)boxp)[bn];
    float4 a = ((const float4*)anch)[n];
    float wa  = a.z - a.x,            ha  = a.w - a.y;
    float cxa = a.x + 0.5f * wa,      cya = a.y + 0.5f * ha;
    float dw  = fminf(d.z, BBOX_CLIP), dh = fminf(d.w, BBOX_CLIP);
    float cx  = d.x * wa + cxa,       cy  = d.y * ha + cya;
    float w   = __expf(dw) * wa,      h   = __expf(dh) * ha;
    float x1 = fminf(fmaxf(cx - 0.5f * w, 0.f), W);
    float y1 = fminf(fmaxf(cy - 0.5f * h, 0.f), H);
    float x2 = fminf(fmaxf(cx + 0.5f * w, 0.f), W);
    float y2 = fminf(fmaxf(cy + 0.5f * h, 0.f), H);
    if ((x2 - x1) < MIN_SIZE || (y2 - y1) < MIN_SIZE) return;

    float sc[4] = {s4.x, s4.y, s4.z, s4.w};
    #pragma unroll
    for (int j = 0; j < 4; ++j) {
        if (sc[j] > SCORE_THRES) {
            int c    = q * 4 + j;
            int slot = atomicAdd(&counts[b * Cn + c], 1);
            if (slot < CAPC) {
                long o = (long)(b * Cn + c) * CAPC + slot;
                cscore[o] = sc[j];
                cidx[o]   = n;
                cbox[o * 4 + 0] = x1; cbox[o * 4 + 1] = y1;
                cbox[o * 4 + 2] = x2; cbox[o * 4 + 3] = y2;
            }
        }
    }
}

// ---------------- kernel 2: per-image sequential class loop ------------------
// One block per image (the reference threads `valid` through classes, forcing
// a sequential class scan). Per class: candidate data is double-buffer
// prefetched into LDS with global_load_async_to_lds (overlapped with the
// previous class's sort/NMS), filtered by the chained valid mask, bitonic
// top-K in LDS, 300x300 suppression-bit matrix (WMMA f32 16x16x4 outer-sum
// computes the pairwise area-sum tiles), bitmask greedy NMS, MAX_DET, output.
__global__ void __launch_bounds__(512) k_nms(
        const int* __restrict__ counts,
        const float* __restrict__ cscore,
        const int* __restrict__ cidx,
        const float* __restrict__ cbox,
        float* __restrict__ out5,
        int* __restrict__ labels,
        float* __restrict__ keepOut) {
    __shared__ unsigned validMask[VWORDS];       // chained valid over N anchors
    __shared__ float    sscore[CAPC];
    __shared__ int      sperm[CAPC];
    __shared__ int      scid[CAPC];
    __shared__ float    sbx[CAPC * 4];
    __shared__ float    tbox[KPAD * 4];
    __shared__ int      tidxA[KPAD];
    __shared__ float    areaL[KPAD];
    __shared__ unsigned supp[KPAD * NWORDS];     // suppression bit matrix
    __shared__ unsigned keepw[NWORDS];
    __shared__ int      scount;
    // async staging buffers (written by the async-DMA engine)
    __shared__ float               gsc[CAPC];
    __shared__ int                 gid[CAPC];
    __shared__ __align__(16) float gbx[CAPC * 4];

    const int b    = blockIdx.x;
    const int tid  = threadIdx.x;
    const int wv   = tid >> 5;        // wave id (16 waves of 32)
    const int lane = tid & 31;
    const int half = lane >> 4;       // WMMA lane group (K=0/1 vs K=2/3)
    const int l15  = lane & 15;

    for (int i = tid; i < VWORDS; i += 512) validMask[i] = 0xFFFFFFFFu;

    // ---- issue async prefetch of class 0 candidate data (fixed CAPC entries,
    //      uniform control flow, EXEC all ones)
    {
        long base0 = (long)(b * Cn) * CAPC;
        async_g2l_b32(&gsc[tid], cscore + base0 + tid);
        async_g2l_b32(&gid[tid], cidx + base0 + tid);
        async_g2l_b128(&gbx[tid * 4], cbox + (base0 + tid) * 4);
    }
    __syncthreads();

    for (int c = 0; c < Cn; ++c) {
        int cnt = counts[b * Cn + c];
        if (cnt > CAPC) cnt = CAPC;
        if (tid == 0) scount = 0;
        wait_async0();                 // staged data for class c is in LDS
        __syncthreads();

        // ---- filter candidates against chained valid mask, compact (from LDS)
        if (tid < cnt) {
            int n = gid[tid];
            if ((validMask[n >> 5] >> (n & 31)) & 1u) {
                int s = atomicAdd(&scount, 1);
                sscore[s] = gsc[tid];
                scid[s]   = n;
                sbx[s * 4 + 0] = gbx[tid * 4 + 0];
                sbx[s * 4 + 1] = gbx[tid * 4 + 1];
                sbx[s * 4 + 2] = gbx[tid * 4 + 2];
                sbx[s * 4 + 3] = gbx[tid * 4 + 3];
            }
        }
        __syncthreads();
        int m = scount;

        // ---- staging free: issue async prefetch for class c+1 (overlaps with
        //      the sort + WMMA + NMS below)
        if (c + 1 < Cn) {
            long basen = (long)(b * Cn + c + 1) * CAPC;
            async_g2l_b32(&gsc[tid], cscore + basen + tid);
            async_g2l_b32(&gid[tid], cidx + basen + tid);
            async_g2l_b128(&gbx[tid * 4], cbox + (basen + tid) * 4);
        }

        // ---- pad + bitonic sort (descending) on (score, perm)
        for (int i = tid; i < CAPC; i += 512) {
            sperm[i] = i;
            if (i >= m) sscore[i] = -3.4e38f;
        }
        __syncthreads();
        for (int size = 2; size <= CAPC; size <<= 1) {
            for (int stride = size >> 1; stride > 0; stride >>= 1) {
                int i = tid;
                int j = i ^ stride;
                if (j > i) {
                    bool desc = ((i & size) == 0);
                    float si = sscore[i], sj = sscore[j];
                    bool sw = desc ? (si < sj) : (si > sj);
                    if (sw) {
                        sscore[i] = sj; sscore[j] = si;
                        int t = sperm[i]; sperm[i] = sperm[j]; sperm[j] = t;
                    }
                }
                __syncthreads();
            }
        }
        int Kp = m < Kk ? m : Kk;

        // ---- gather top-K boxes, areas; clear suppression matrix
        for (int k = tid; k < KPAD; k += 512) {
            if (k < Kp) {
                int s = sperm[k];
                float bx0 = sbx[s * 4 + 0], bx1 = sbx[s * 4 + 1];
                float bx2 = sbx[s * 4 + 2], bx3 = sbx[s * 4 + 3];
                tbox[k * 4 + 0] = bx0; tbox[k * 4 + 1] = bx1;
                tbox[k * 4 + 2] = bx2; tbox[k * 4 + 3] = bx3;
                tidxA[k] = scid[s];
                areaL[k] = (bx2 - bx0) * (bx3 - bx1);
            } else {
                tbox[k * 4 + 0] = 0.f; tbox[k * 4 + 1] = 0.f;
                tbox[k * 4 + 2] = 0.f; tbox[k * 4 + 3] = 0.f;
                tidxA[k] = 0;
                areaL[k] = 0.f;
            }
        }
        for (int i = tid; i < KPAD * NWORDS; i += 512) supp[i] = 0u;
        __syncthreads();

        // ---- pairwise suppression matrix, 16x16 tiles per wave.
        // WMMA outer-sum: A=[area_i, 1, 0, 0] (16x4), B=[1; area_j; 0; 0]
        // (4x16)  =>  D[m][n] = area_i[m] + area_j[n]  (IoU denominator base).
        int nT = (Kp + 15) >> 4;                 // block-uniform
        for (int t = wv; t < nT * nT; t += 16) { // wave-uniform loop
            int ti = t / nT, tj = t % nT;
            v2f A, Bv;
            float ai = areaL[ti * 16 + l15];
            float aj = areaL[tj * 16 + l15];
            A.x  = half ? 0.f : ai;   A.y  = half ? 0.f : 1.f;   // K=0,1 | K=2,3->0
            Bv.x = half ? 0.f : 1.f;  Bv.y = half ? 0.f : aj;
            v8f Cz = {0.f, 0.f, 0.f, 0.f, 0.f, 0.f, 0.f, 0.f};
            v8f S = __builtin_amdgcn_wmma_f32_16x16x4_f32(
                        false, A, false, Bv, (short)0, Cz, false, false);
            #pragma unroll
            for (int r = 0; r < 8; ++r) {
                int mrow = r + (half << 3);      // C/D layout: lanes16-31 => M+8
                int gi = ti * 16 + mrow;
                int gj = tj * 16 + l15;
                if (gi < Kp && gj < Kp && gj > gi) {
                    const float* bi = &tbox[gi * 4];
                    const float* bj = &tbox[gj * 4];
                    float xx1 = fmaxf(bi[0], bj[0]), yy1 = fmaxf(bi[1], bj[1]);
                    float xx2 = fminf(bi[2], bj[2]), yy2 = fminf(bi[3], bj[3]);
                    float iw = fmaxf(xx2 - xx1, 0.f), ih = fmaxf(yy2 - yy1, 0.f);
                    float inter = iw * ih;
                    float uni = S[r] - inter + 1e-9f;   // area_i+area_j-inter
                    if (inter > NMS_THRES * uni)
                        atomicOr(&supp[gi * NWORDS + (gj >> 5)], 1u << (gj & 31));
                }
            }
        }
        __syncthreads();

        // ---- sequential greedy scan over bitmask (tiny: <=300 iters)
        if (tid == 0) {
            int nw = (Kp + 31) >> 5;
            unsigned kw[NWORDS];
            for (int w0 = 0; w0 < NWORDS; ++w0) {
                int lo = w0 * 32;
                unsigned v = 0u;
                if (lo + 32 <= Kp)      v = 0xFFFFFFFFu;
                else if (lo < Kp)       v = (1u << (Kp - lo)) - 1u;
                kw[w0] = v;
            }
            for (int i = 0; i < Kp; ++i)
                if ((kw[i >> 5] >> (i & 31)) & 1u)
                    for (int w0 = 0; w0 < nw; ++w0)
                        kw[w0] &= ~supp[i * NWORDS + w0];
            int kept = 0;                      // MAX_DET truncation (post-NMS)
            for (int i = 0; i < Kp; ++i)
                if ((kw[i >> 5] >> (i & 31)) & 1u) {
                    if (++kept > MAXDET) kw[i >> 5] &= ~(1u << (i & 31));
                }
            for (int w0 = 0; w0 < NWORDS; ++w0) keepw[w0] = kw[w0];
        }
        __syncthreads();

        // ---- new_valid = zeros.at[idx].set(keep)
        for (int i = tid; i < VWORDS; i += 512) validMask[i] = 0u;
        __syncthreads();
        if (tid < Kp && ((keepw[tid >> 5] >> (tid & 31)) & 1u)) {
            int n = tidxA[tid];
            atomicOr(&validMask[n >> 5], 1u << (n & 31));
        }

        // ---- outputs for (b, c)
        long ob = (long)(b * Cn + c) * Kk;
        for (int k = tid; k < Kk; k += 512) {
            bool kp = (k < Kp) && ((keepw[k >> 5] >> (k & 31)) & 1u);
            float* o = &out5[(ob + k) * 5];
            if (kp) {
                o[0] = tbox[k * 4 + 0]; o[1] = tbox[k * 4 + 1];
                o[2] = tbox[k * 4 + 2]; o[3] = tbox[k * 4 + 3];
                o[4] = sscore[k];
            } else {
                o[0] = 0.f; o[1] = 0.f; o[2] = 0.f; o[3] = 0.f; o[4] = 0.f;
            }
            labels[ob + k]  = c;
            keepOut[ob + k] = kp ? 1.f : 0.f;
        }
        __syncthreads();
    }
}

// ---------------- launch ----------------
extern "C" void kernel_launch(void* const* d_in, const int* in_sizes, int n_in,
                              void* d_out, int out_size, void* d_ws, size_t ws_size,
                              hipStream_t stream) {
    const float* boxp = (const float*)d_in[0];   // (B,N,4) f32
    const float* clsp = (const float*)d_in[1];   // (B,N,C) f32
    const float* anch = (const float*)d_in[2];   // (N,4)   f32
    const int*   Hs   = (const int*)d_in[3];     // scalar
    const int*   Ws   = (const int*)d_in[4];     // scalar

    // workspace layout (16B-aligned segments)
    int*   counts = (int*)d_ws;                                  // B*C
    float* cscore = (float*)((char*)d_ws + 1024);                // B*C*CAPC
    int*   cidx   = (int*)(cscore + (size_t)Bn * Cn * CAPC);
    float* cbox   = (float*)(cidx + (size_t)Bn * Cn * CAPC);     // B*C*CAPC*4

    // output layout: out5 | labels | keep
    float* out5    = (float*)d_out;
    int*   labels  = (int*)(out5 + (size_t)Bn * Cn * Kk * 5);
    float* keepOut = (float*)(labels + (size_t)Bn * Cn * Kk);

    k_zero_counts<<<1, 256, 0, stream>>>(counts);

    long total = (long)Bn * Nn * (Cn / 4);
    int  blocks = (int)((total + 255) / 256);
    k_gather<<<blocks, 256, 0, stream>>>(boxp, clsp, anch, Hs, Ws,
                                         counts, cscore, cidx, cbox);

    k_nms<<<Bn, 512, 0, stream>>>(counts, cscore, cidx, cbox,
                                  out5, labels, keepOut);
}